// P4_60413009986089
// MI455X (gfx1250) — compile-verified
//
#include <hip/hip_runtime.h>
#include <stdint.h>

#define BATCH   16
#define NANCH   25200
#define NCLS    80
#define ROWSZ   85
#define TOPK    1000
#define MAXDET  300
#define CONF_T  0.3f
#define IOU_T   0.45f

// ---- CDNA5 async global->LDS helpers (gfx1250) --------------------------
// Low 32 bits of a flat shared-aperture address == LDS byte offset (ISA 10.2).
__device__ __forceinline__ unsigned lds_lo32(const void* p) {
  return (unsigned)(uintptr_t)p;
}

// GLOBAL_LOAD_ASYNC_TO_LDS_B128, GVS mode: saddr(64b SGPR) + vaddr(32b byte off),
// vdst = per-lane LDS byte address. Tracked by ASYNCcnt.
__device__ __forceinline__ void async_b128(unsigned lds_addr, unsigned byte_off,
                                           const void* sbase) {
  asm volatile("global_load_async_to_lds_b128 %0, %1, %2"
               :
               : "v"(lds_addr), "v"(byte_off), "s"(sbase)
               : "memory");
}

__device__ __forceinline__ void wait_async0() {
  asm volatile("s_wait_asynccnt 0" ::: "memory");
}

// ---- Kernel 1: per-anchor score/class/key (bandwidth-bound stage) -------
// 64 anchors per 256-thread block; tile of 64*85 floats staged via async TDM path.
__global__ __launch_bounds__(256) void score_kernel(
    const float* __restrict__ pred, unsigned* __restrict__ keys,
    int* __restrict__ clsArr) {
  __shared__ __align__(16) float tile[64 * ROWSZ];  // 5440 floats = 1360 x b128
  const int tid = threadIdx.x;
  const int blk = blockIdx.x;
  const float* gbase = pred + (size_t)blk * (64 * ROWSZ);
  const unsigned lbase = lds_lo32(tile);

#pragma unroll
  for (int it = 0; it < 6; ++it) {
    int c = it * 256 + tid;
    if (c < 1360) async_b128(lbase + (unsigned)c * 16u, (unsigned)c * 16u, gbase);
  }
  wait_async0();
  __syncthreads();

  const int la = tid >> 2;      // local anchor 0..63
  const int part = tid & 3;     // 4 lanes per anchor, 20 classes each
  const float* row = &tile[la * ROWSZ];
  const float obj = row[4];

  float m = -1.0f;
  int mi = 0;
#pragma unroll
  for (int k = 0; k < 20; ++k) {
    int c = part * 20 + k;
    float v = row[5 + c];
    if (v > m) { m = v; mi = c; }
  }
  // combine 4 partials within the wave (wave32, groups of 4 lanes aligned)
#pragma unroll
  for (int d = 1; d < 4; d <<= 1) {
    float om = __shfl_xor(m, d, 32);
    int oi = __shfl_xor(mi, d, 32);
    if (om > m || (om == m && oi < mi)) { m = om; mi = oi; }
  }

  if (part == 0) {
    float score = m * obj;  // max(cls)*obj == max(cls*obj) for obj>=0
    bool valid = (obj > CONF_T) && (score > CONF_T);
    int ga = blk * 64 + la;
    keys[ga] = valid ? __float_as_uint(score) : 0u;  // score>0.3 => positive-float bit order
    clsArr[ga] = mi;
  }
}

// ---- Kernel 2: per-batch top-1000 (LDS-resident iterative argmax) -------
__global__ __launch_bounds__(1024) void topk_kernel(
    const unsigned* __restrict__ keys, int* __restrict__ topIdx,
    unsigned* __restrict__ topKey) {
  extern __shared__ __align__(16) unsigned char smem2[];
  unsigned long long* warpRes = (unsigned long long*)smem2;          // 32 * 8B
  unsigned long long* sBest   = (unsigned long long*)(smem2 + 256);  // 8B
  unsigned* skeys             = (unsigned*)(smem2 + 272);            // 25200 * 4B

  const int tid = threadIdx.x;
  const int b = blockIdx.x;
  const unsigned* kb = keys + (size_t)b * NANCH;
  const unsigned lbase = lds_lo32(skeys);

  // 25200 dwords = 6300 b128 chunks, async staged to LDS
  for (int it = 0; it < 7; ++it) {
    int c = it * 1024 + tid;
    if (c < 6300) async_b128(lbase + (unsigned)c * 16u, (unsigned)c * 16u, kb);
  }
  wait_async0();
  __syncthreads();

  // cached per-thread local max over strided slice (<=25 entries)
  unsigned long long lb = 0ull;
  for (int idx = tid; idx < NANCH; idx += 1024) {
    unsigned long long p = ((unsigned long long)skeys[idx] << 32) |
                           (unsigned long long)(0xFFFFFFFFu - (unsigned)idx);
    if (p > lb) lb = p;
  }

  int* tI = topIdx + (size_t)b * TOPK;
  unsigned* tK = topKey + (size_t)b * TOPK;

  for (int k = 0; k < TOPK; ++k) {
    unsigned long long v = lb;
#pragma unroll
    for (int d = 16; d > 0; d >>= 1) {
      unsigned long long o = __shfl_down(v, (unsigned)d, 32);
      if (o > v) v = o;
    }
    if ((tid & 31) == 0) warpRes[tid >> 5] = v;
    __syncthreads();
    if (tid < 32) {
      unsigned long long x = warpRes[tid];
#pragma unroll
      for (int d = 16; d > 0; d >>= 1) {
        unsigned long long o = __shfl_down(x, (unsigned)d, 32);
        if (o > x) x = o;
      }
      if (tid == 0) *sBest = x;
    }
    __syncthreads();
    const unsigned long long best = *sBest;
    const unsigned bk = (unsigned)(best >> 32);
    const int bi = (int)(0xFFFFFFFFu - (unsigned)(best & 0xFFFFFFFFull));

    if (bk == 0u) {  // uniform: no valid candidates remain
      for (int kk = k + tid; kk < TOPK; kk += 1024) { tI[kk] = 0; tK[kk] = 0u; }
      break;
    }
    if (tid == 0) { tI[k] = bi; tK[k] = bk; }
    if ((bi & 1023) == tid) {  // owner removes + rescans its slice only
      skeys[bi] = 0u;
      unsigned long long nb = 0ull;
      for (int idx = tid; idx < NANCH; idx += 1024) {
        unsigned long long p = ((unsigned long long)skeys[idx] << 32) |
                               (unsigned long long)(0xFFFFFFFFu - (unsigned)idx);
        if (p > nb) nb = p;
      }
      lb = nb;
    }
    __syncthreads();
  }
}

// ---- Kernel 3: greedy NMS scan + emit top-300 ---------------------------
__global__ __launch_bounds__(1024) void nms_kernel(
    const float* __restrict__ pred, const int* __restrict__ clsArr,
    const int* __restrict__ topIdx, const unsigned* __restrict__ topKey,
    float* __restrict__ out) {
  __shared__ float bx1[TOPK], by1[TOPK], bx2[TOPK], by2[TOPK];
  __shared__ float ba[TOPK], bs[TOPK];
  __shared__ int bcls[TOPK];
  __shared__ unsigned kmask[TOPK / 32], vmask[TOPK / 32];
  __shared__ int flag;

  const int tid = threadIdx.x;
  const int b = blockIdx.x;

  if (tid < TOPK / 32) { kmask[tid] = 0u; vmask[tid] = 0u; }
  __syncthreads();

  if (tid < TOPK) {
    const unsigned key = topKey[(size_t)b * TOPK + tid];
    const int idx = topIdx[(size_t)b * TOPK + tid];
    const float* r = pred + ((size_t)b * NANCH + (size_t)idx) * ROWSZ;
    const float x = r[0], y = r[1], w = r[2], h = r[3];
    const float x1 = x - w * 0.5f, y1 = y - h * 0.5f;
    const float x2 = x + w * 0.5f, y2 = y + h * 0.5f;
    bx1[tid] = x1; by1[tid] = y1; bx2[tid] = x2; by2[tid] = y2;
    ba[tid] = (x2 - x1) * (y2 - y1);
    bs[tid] = __uint_as_float(key);
    bcls[tid] = clsArr[(size_t)b * NANCH + (size_t)idx];
    if (key != 0u) atomicOr(&vmask[tid >> 5], 1u << (tid & 31));
  }
  __syncthreads();

  // sequential greedy scan; each step parallel over previously-kept j.
  // class inequality <=> zero IoU under the reference's +cls*4096 offset.
  for (int i = 0; i < TOPK; ++i) {
    if (tid == 0) flag = 0;
    __syncthreads();
    const int j = tid;
    if (j < i && ((kmask[j >> 5] >> (j & 31)) & 1u) && bcls[j] == bcls[i]) {
      float xx1 = fmaxf(bx1[i], bx1[j]);
      float yy1 = fmaxf(by1[i], by1[j]);
      float xx2 = fminf(bx2[i], bx2[j]);
      float yy2 = fminf(by2[i], by2[j]);
      float ww = fmaxf(xx2 - xx1, 0.0f), hh = fmaxf(yy2 - yy1, 0.0f);
      float inter = ww * hh;
      float iou = inter / (ba[i] + ba[j] - inter + 1e-7f);
      if (iou > IOU_T) flag = 1;  // benign race, all write 1
    }
    __syncthreads();
    if (tid == 0) {
      bool vi = (vmask[i >> 5] >> (i & 31)) & 1u;
      if (vi && !flag) kmask[i >> 5] |= 1u << (i & 31);
    }
    __syncthreads();
  }

  // kept entries are already score-descending (top-k order) -> emit in order
  if (tid == 0) {
    float* ob = out + (size_t)b * MAXDET * 6;
    int cnt = 0;
    for (int k = 0; k < TOPK && cnt < MAXDET; ++k) {
      if ((kmask[k >> 5] >> (k & 31)) & 1u) {
        float* o = ob + cnt * 6;
        o[0] = bx1[k]; o[1] = by1[k]; o[2] = bx2[k]; o[3] = by2[k];
        o[4] = bs[k];  o[5] = (float)bcls[k];
        ++cnt;
      }
    }
    for (; cnt < MAXDET; ++cnt) {
      float* o = ob + cnt * 6;
      o[0] = 0.0f; o[1] = 0.0f; o[2] = 0.0f; o[3] = 0.0f; o[4] = 0.0f; o[5] = 0.0f;
    }
  }
}

extern "C" void kernel_launch(void* const* d_in, const int* in_sizes, int n_in,
                              void* d_out, int out_size, void* d_ws, size_t ws_size,
                              hipStream_t stream) {
  const float* pred = (const float*)d_in[0];
  float* out = (float*)d_out;

  // workspace layout (~3.35 MB)
  unsigned char* w = (unsigned char*)d_ws;
  unsigned* keys  = (unsigned*)(w);
  int* clsArr     = (int*)(w + (size_t)BATCH * NANCH * 4);
  int* topIdx     = (int*)(w + 2 * (size_t)BATCH * NANCH * 4);
  unsigned* topKey = (unsigned*)(w + 2 * (size_t)BATCH * NANCH * 4 +
                                 (size_t)BATCH * TOPK * 4);

  // Stage 1: 403200 anchors / 64 per block = 6300 blocks
  score_kernel<<<(BATCH * NANCH) / 64, 256, 0, stream>>>(pred, keys, clsArr);

  // Stage 2: one WGP-resident block per batch, 100.9 KB dynamic LDS (<=320KB/WGP)
  const size_t sh2 = 272 + (size_t)NANCH * 4;
  (void)hipFuncSetAttribute((const void*)topk_kernel,
                            hipFuncAttributeMaxDynamicSharedMemorySize, (int)sh2);
  topk_kernel<<<BATCH, 1024, sh2, stream>>>(keys, topIdx, topKey);

  // Stage 3: greedy NMS + emit
  nms_kernel<<<BATCH, 1024, 0, stream>>>(pred, clsArr, topIdx, topKey, out);
}